// DCRNN_90520730730510
// MI455X (gfx1250) — compile-verified
//
#include <hip/hip_runtime.h>
#include <hip/hip_bf16.h>
#include <math.h>

#define N_NODES 50000
#define HID     64
#define LAYERS  2
#define STEPS   12
#define NEDGE   800000

typedef float v2f __attribute__((ext_vector_type(2)));
typedef float v8f __attribute__((ext_vector_type(8)));

__device__ __forceinline__ float sigmoidf_(float x) { return 1.0f / (1.0f + __expf(-x)); }

__device__ __forceinline__ void atomicAddF(float* p, float v) {
    unsafeAtomicAdd(p, v);   // lowers to global_atomic_add_f32 (no CAS loop)
}

// ---------------- degree / normalization precompute ----------------

__global__ void deg_kernel(const long long* __restrict__ src,
                           const long long* __restrict__ dst,
                           float* __restrict__ degf,   // counts of dst
                           float* __restrict__ degb,   // counts of src
                           int E)
{
    int e = blockIdx.x * blockDim.x + threadIdx.x;
    if (e >= E) return;
    atomicAddF(&degf[(int)dst[e]], 1.0f);
    atomicAddF(&degb[(int)src[e]], 1.0f);
}

__global__ void dinv_kernel(float* __restrict__ dinvf, float* __restrict__ dinvb,
                            float* __restrict__ selff, float* __restrict__ selfb, int N)
{
    int n = blockIdx.x * blockDim.x + threadIdx.x;
    if (n >= N) return;
    float df = dinvf[n] + 1.0f;   // +1 self loop
    float db = dinvb[n] + 1.0f;
    dinvf[n] = rsqrtf(df);
    dinvb[n] = rsqrtf(db);
    selff[n] = 1.0f / df;
    selfb[n] = 1.0f / db;
}

__global__ void coef_kernel(const long long* __restrict__ src,
                            const long long* __restrict__ dst,
                            const float* __restrict__ dinvf,
                            const float* __restrict__ dinvb,
                            float* __restrict__ coef_f,
                            float* __restrict__ coef_b, int E)
{
    int e = blockIdx.x * blockDim.x + threadIdx.x;
    if (e >= E) return;
    int s = (int)src[e], d = (int)dst[e];
    coef_f[e] = dinvf[s] * dinvf[d];
    coef_b[e] = dinvb[s] * dinvb[d];
}

// ---------------- input projection: cur = x_t @ proj_w + proj_b ----------------

__global__ void proj_kernel(const float* __restrict__ xt,   // [N,2]
                            const float* __restrict__ pw,   // [2,64]
                            const float* __restrict__ pb,   // [64]
                            float* __restrict__ cur, int N)
{
    int idx = blockIdx.x * blockDim.x + threadIdx.x;
    if (idx >= N * HID) return;
    int n = idx >> 6, f = idx & 63;
    cur[idx] = xt[n * 2 + 0] * pw[f] + xt[n * 2 + 1] * pw[HID + f] + pb[f];
}

// ---------------- WMMA f32 GEMM: D[N x MOUT] = A[N x 64] @ B ----------------
// B element (k,m) lives at B[k*bKs + m*bMs].  One wave = 16 rows x MOUT cols.
// LDS holds B in K-pair-interleaved form: (k,m) -> lds[(k>>1)*PITCH + 2*m + (k&1)],
// so each lane's {B[k,m], B[k+1,m]} is one contiguous 8B ds_load_b64 into an
// even-aligned VGPR pair.  PITCH mod 64 == 32 -> lanes 0-15 and 16-31 hit
// complementary halves of the 64 LDS banks (conflict-free).

template <int MOUT>
__global__ __launch_bounds__(128)
void gemm_wmma(const float* __restrict__ A, const float* __restrict__ B,
               int bKs, int bMs, float* __restrict__ D, int ntiles)
{
    constexpr int PITCH = 2 * MOUT + 32;     // floats per k-pair row
    __shared__ float ldsB[32 * PITCH];

    const int tid = threadIdx.x;
    for (int i = tid; i < 64 * MOUT; i += 128) {
        int k = i / MOUT;
        int m = i - k * MOUT;
        ldsB[(k >> 1) * PITCH + 2 * m + (k & 1)] = B[k * bKs + m * bMs];
    }
    __syncthreads();

    const int wid  = tid >> 5;
    const int lane = tid & 31;
    const int tile = blockIdx.x * 4 + wid;
    if (tile >= ntiles) return;             // uniform per wave; after the only barrier

    const int lr = lane & 15;               // M (A rows) / N (B cols) index
    const int hk = lane >> 4;               // K half-select

    v8f acc[MOUT / 16];
    const v8f vzero = {0.f, 0.f, 0.f, 0.f, 0.f, 0.f, 0.f, 0.f};
#pragma unroll
    for (int j = 0; j < MOUT / 16; ++j) acc[j] = vzero;

    const float* Arow = A + (size_t)(tile * 16 + lr) * 64;

    for (int k0 = 0; k0 < 64; k0 += 4) {
        const int kb = k0 + hk * 2;
        float2 av = *reinterpret_cast<const float2*>(Arow + kb);
        v2f a; a[0] = av.x; a[1] = av.y;
        const float* brow = &ldsB[(kb >> 1) * PITCH + 2 * lr];
#pragma unroll
        for (int j = 0; j < MOUT / 16; ++j) {
            v2f b = *reinterpret_cast<const v2f*>(brow + j * 32);
            acc[j] = __builtin_amdgcn_wmma_f32_16x16x4_f32(
                false, a, false, b, (short)0, acc[j], false, false);
        }
    }

    float* Dtile = D + (size_t)tile * 16 * MOUT;
#pragma unroll
    for (int j = 0; j < MOUT / 16; ++j) {
#pragma unroll
        for (int v = 0; v < 8; ++v) {
            Dtile[(size_t)(v + 8 * hk) * MOUT + j * 16 + lr] = acc[j][v];
        }
    }
}

// ---------------- edge scatter: agg[to] += h[from] * coef ----------------

__global__ void scatter_kernel(const long long* __restrict__ from_idx,
                               const long long* __restrict__ to_idx,
                               const float* __restrict__ h,
                               const float* __restrict__ coef,
                               float* __restrict__ agg, int E)
{
    int t = blockIdx.x * blockDim.x + threadIdx.x;
    if (t >= E * HID) return;
    int e = t >> 6, f = t & 63;
    int a = (int)from_idx[e];
    int b = (int)to_idx[e];
    atomicAddF(&agg[(size_t)b * HID + f], h[(size_t)a * HID + f] * coef[e]);
}

// ---------------- combine: cur = relu(cf + cb) ----------------

__global__ void combine_kernel(const float* __restrict__ aggf, const float* __restrict__ aggb,
                               const float* __restrict__ hf,   const float* __restrict__ hb,
                               const float* __restrict__ selff, const float* __restrict__ selfb,
                               const float* __restrict__ bf,   const float* __restrict__ bb,
                               float* __restrict__ cur, int N)
{
    int idx = blockIdx.x * blockDim.x + threadIdx.x;
    if (idx >= N * HID) return;
    int n = idx >> 6, f = idx & 63;
    float v = aggf[idx] + hf[idx] * selff[n] + bf[f]
            + aggb[idx] + hb[idx] * selfb[n] + bb[f];
    cur[idx] = fmaxf(v, 0.0f);
}

// ---------------- GRU cell (torch gate order r,z,n) ----------------

__global__ void gru_kernel(const float* __restrict__ gi,   // [N,192] (no bias)
                           const float* __restrict__ gh,   // [N,192] (no bias)
                           const float* __restrict__ bih,  // [192]
                           const float* __restrict__ bhh,  // [192]
                           float* __restrict__ h,          // [N,64] in/out
                           float* __restrict__ cur,        // [N,64] out (next layer input)
                           int N)
{
    int idx = blockIdx.x * blockDim.x + threadIdx.x;
    if (idx >= N * HID) return;
    int n = idx >> 6, f = idx & 63;
    const float* gin = gi + (size_t)n * 192;
    const float* ghn = gh + (size_t)n * 192;
    float r  = sigmoidf_((gin[f]           + bih[f])
                       + (ghn[f]           + bhh[f]));
    float z  = sigmoidf_((gin[HID + f]     + bih[HID + f])
                       + (ghn[HID + f]     + bhh[HID + f]));
    float nn = tanhf((gin[2 * HID + f] + bih[2 * HID + f])
                   + r * (ghn[2 * HID + f] + bhh[2 * HID + f]));
    float hv = h[idx];
    float hnew = (1.0f - z) * nn + z * hv;
    h[idx]   = hnew;
    cur[idx] = hnew;
}

// ---------------- output head: out[n] = h1[n,:] . out_w + out_b ----------------

__global__ void out_kernel(const float* __restrict__ h1,
                           const float* __restrict__ ow,
                           const float* __restrict__ ob,
                           float* __restrict__ out, int N)
{
    int n = blockIdx.x * blockDim.x + threadIdx.x;
    if (n >= N) return;
    const float* hr = h1 + (size_t)n * HID;
    float s = 0.0f;
#pragma unroll
    for (int k = 0; k < HID; ++k) s += hr[k] * ow[k];
    out[n] = s + ob[0];
}

// ---------------- host driver ----------------

extern "C" void kernel_launch(void* const* d_in, const int* in_sizes, int n_in,
                              void* d_out, int out_size, void* d_ws, size_t ws_size,
                              hipStream_t stream)
{
    const float*     x       = (const float*)d_in[0];
    const long long* ei      = (const long long*)d_in[1];
    const float*     proj_w  = (const float*)d_in[2];
    const float*     proj_b  = (const float*)d_in[3];
    const float*     convf_w = (const float*)d_in[4];
    const float*     convf_b = (const float*)d_in[5];
    const float*     convb_w = (const float*)d_in[6];
    const float*     convb_b = (const float*)d_in[7];
    const float*     gru_wih = (const float*)d_in[8];
    const float*     gru_bih = (const float*)d_in[9];
    const float*     gru_whh = (const float*)d_in[10];
    const float*     gru_bhh = (const float*)d_in[11];
    const float*     out_w   = (const float*)d_in[12];
    const float*     out_b   = (const float*)d_in[13];
    float*           out     = (float*)d_out;

    const int N = N_NODES, E = NEDGE;
    const size_t NH = (size_t)N * HID;

    float* w      = (float*)d_ws;
    float* dinvf  = w;                       // [N] (counts, then rsqrt)
    float* dinvb  = dinvf + N;               // [N]
    float* selff  = dinvb + N;               // [N]
    float* selfb  = selff + N;               // [N]
    float* coef_f = selfb + N;               // [E]
    float* coef_b = coef_f + E;              // [E]
    float* h0     = coef_b + E;              // [N,64]
    float* h1     = h0 + NH;                 // [N,64]
    float* cur    = h1 + NH;                 // [N,64]
    float* scr    = cur + NH;                // [N, 384] scratch union
    float* hf     = scr;                     // [N,64]
    float* hb     = scr + NH;                // [N,64]
    float* aggf   = scr + NH * 2;            // [N,64]
    float* aggb   = scr + NH * 3;            // [N,64]
    float* gi     = scr;                     // [N,192] (aliases hf/hb/aggf, dead by then)
    float* gh     = scr + (size_t)N * 192;   // [N,192] (aliases aggb)

    const long long* src = ei;
    const long long* dst = ei + E;

    // ---- prologue: degrees, norms, coefs, h=0 ----
    hipMemsetAsync(dinvf, 0, 2 * N * sizeof(float), stream);           // degf|degb
    hipMemsetAsync(h0, 0, 2 * NH * sizeof(float), stream);             // h0|h1
    deg_kernel <<<(E + 255) / 256, 256, 0, stream>>>(src, dst, dinvf, dinvb, E);
    dinv_kernel<<<(N + 255) / 256, 256, 0, stream>>>(dinvf, dinvb, selff, selfb, N);
    coef_kernel<<<(E + 255) / 256, 256, 0, stream>>>(src, dst, dinvf, dinvb, coef_f, coef_b, E);

    const int ntiles = N / 16;                       // 3125 (exact)
    const int gemm_blocks = (ntiles + 3) / 4;        // 4 waves / block
    const int ew_blocks = (N * HID + 255) / 256;
    const int sc_blocks = (E * HID + 255) / 256;
    float* hlay[2] = {h0, h1};

    for (int t = 0; t < STEPS; ++t) {
        proj_kernel<<<ew_blocks, 256, 0, stream>>>(x + (size_t)t * N * 2, proj_w, proj_b, cur, N);
        for (int l = 0; l < LAYERS; ++l) {
            // diffusion convs: hf = cur@Wf, hb = cur@Wb   (B row-major [64][64])
            gemm_wmma<64><<<gemm_blocks, 128, 0, stream>>>(
                cur, convf_w + (size_t)l * 64 * 64, 64, 1, hf, ntiles);
            gemm_wmma<64><<<gemm_blocks, 128, 0, stream>>>(
                cur, convb_w + (size_t)l * 64 * 64, 64, 1, hb, ntiles);
            hipMemsetAsync(aggf, 0, 2 * NH * sizeof(float), stream);   // aggf|aggb
            scatter_kernel<<<sc_blocks, 256, 0, stream>>>(src, dst, hf, coef_f, aggf, E);
            scatter_kernel<<<sc_blocks, 256, 0, stream>>>(dst, src, hb, coef_b, aggb, E);
            combine_kernel<<<ew_blocks, 256, 0, stream>>>(
                aggf, aggb, hf, hb, selff, selfb,
                convf_b + l * HID, convb_b + l * HID, cur, N);
            // GRU gates: gi = cur @ Wih^T, gh = h @ Whh^T   (B(k,m) = W[m*64+k])
            gemm_wmma<192><<<gemm_blocks, 128, 0, stream>>>(
                cur, gru_wih + (size_t)l * 192 * 64, 1, 64, gi, ntiles);
            gemm_wmma<192><<<gemm_blocks, 128, 0, stream>>>(
                hlay[l], gru_whh + (size_t)l * 192 * 64, 1, 64, gh, ntiles);
            gru_kernel<<<ew_blocks, 256, 0, stream>>>(
                gi, gh, gru_bih + l * 192, gru_bhh + l * 192, hlay[l], cur, N);
        }
    }
    out_kernel<<<(N + 255) / 256, 256, 0, stream>>>(h1, out_w, out_b, out, N);
}